// nn_layer_47081431498926
// MI455X (gfx1250) — compile-verified
//
#include <hip/hip_runtime.h>
#include <cstdint>

// ---------------------------------------------------------------------------
// MI455X (gfx1250) kernel: per-row 3-way top-16 selection + gather.
//   traffic floor: d_lon + d_lat = 1.07 GB read once  ->  ~46 us @ 23.3 TB/s
//   No matmul in this op => WMMA inapplicable. CDNA5 path exercised instead:
//   double-buffered GLOBAL_LOAD_ASYNC_TO_LDS_B128 staging + s_wait_asynccnt,
//   compute reads tiles from LDS (ds_load_b128), wave32 shfl merges.
// ---------------------------------------------------------------------------

namespace {

constexpr int S_DIM  = 16384;            // columns per row
constexpr int T_DIM  = 8192;             // rows
constexpr int B_DIM  = 16;               // batch
constexpr int NH     = 16;               // top-k
constexpr int TILE   = 512;              // floats per tile per array (2 KB)
constexpr int NTILES = S_DIM / TILE;     // 32
constexpr int WPB    = 8;                // waves per block
constexpr int BLOCK  = WPB * 32;         // 256 threads
constexpr int CHUNKS = (TILE * 4) / 512; // 4 x 512B async b128 issues per array

// flat float offsets in d_out (tuple concatenated in return order)
constexpr size_t XN_OFF    = 0;
constexpr size_t IDD_OFF   = (size_t)B_DIM * T_DIM * NH;   // 2097152
constexpr size_t IDLON_OFF = IDD_OFF + (size_t)T_DIM * NH; // 2228224
constexpr size_t IDLAT_OFF = IDLON_OFF + (size_t)T_DIM * NH;

// --- CDNA5 async global->LDS copy (32 lanes x 16B = 512B per issue) ---------
__device__ __forceinline__ void async_b128(uint32_t lds_addr, uint64_t gaddr) {
  asm volatile("global_load_async_to_lds_b128 %0, %1, off"
               :: "v"(lds_addr), "v"(gaddr) : "memory");
}
__device__ __forceinline__ void wait_async_le8() {
  asm volatile("s_wait_asynccnt 8" ::: "memory");
}
__device__ __forceinline__ void wait_async_0() {
  asm volatile("s_wait_asynccnt 0" ::: "memory");
}
__device__ __forceinline__ void wait_ds_0() {
  asm volatile("s_wait_dscnt 0" ::: "memory");
}

__device__ __forceinline__ float wave_min32(float x) {
#pragma unroll
  for (int off = 16; off > 0; off >>= 1) x = fminf(x, __shfl_xor(x, off, 32));
  return x;
}

// sorted-ascending 16-entry register list; strict '<' keeps within-lane
// (monotonically increasing index) tie-breaking exact vs jax top_k.
__device__ __forceinline__ void insert16(float key, int idx,
                                         float (&v)[16], int (&id)[16]) {
  if (key < v[15]) {
    v[15] = key; id[15] = idx;
#pragma unroll
    for (int j = 15; j > 0; --j) {
      if (v[j] < v[j - 1]) {
        float tv = v[j]; v[j] = v[j - 1]; v[j - 1] = tv;
        int   ti = id[j]; id[j] = id[j - 1]; id[j - 1] = ti;
      }
    }
  }
}

} // namespace

__global__ __launch_bounds__(BLOCK) void knn_topk_gather_gfx1250(
    const float* __restrict__ x,      // [16, 16384]
    const float* __restrict__ dlon,   // [8192, 16384]
    const float* __restrict__ dlat,   // [8192, 16384]
    float* __restrict__ out)
{
  // per wave: 2 buffers x (TILE lon + TILE lat) = 8 KB; 64 KB per block
  __shared__ __align__(16) float smem[WPB * 2 * 2 * TILE];

  const int lane = (int)(threadIdx.x & 31);
  const int wave = (int)(threadIdx.x >> 5);
  const int row  = (int)blockIdx.x * WPB + wave;   // one wave per row

  const float* glon = dlon + (size_t)row * S_DIM;
  const float* glat = dlat + (size_t)row * S_DIM;

  float* wbase = &smem[wave * (2 * 2 * TILE)];
  // low 32 bits of a generic LDS pointer == workgroup-relative LDS byte addr
  const uint32_t lds_wbase = (uint32_t)(uintptr_t)wbase;

  auto issue_tile = [&](int tile, int buf) {
    const uint32_t lb = lds_wbase + (uint32_t)(buf * 2 * TILE * 4);
    const uint64_t gl = (uint64_t)(uintptr_t)(glon + tile * TILE);
    const uint64_t ga = (uint64_t)(uintptr_t)(glat + tile * TILE);
    const uint32_t lo = (uint32_t)(lane * 16);
#pragma unroll
    for (int c = 0; c < CHUNKS; ++c)
      async_b128(lb + c * 512 + lo, gl + c * 512 + lo);
#pragma unroll
    for (int c = 0; c < CHUNKS; ++c)
      async_b128(lb + TILE * 4 + c * 512 + lo, ga + c * 512 + lo);
  };

  // three per-lane top-16 lists: dist, |lon|, |lat|
  float vd[16], vl[16], va[16];
  int   jd[16], jl[16], ja[16];
#pragma unroll
  for (int i = 0; i < 16; ++i) {
    vd[i] = vl[i] = va[i] = __builtin_inff();
    jd[i] = jl[i] = ja[i] = 0x7fffffff;
  }
  // wave-uniform conservative thresholds (>= true global 16th-best seen)
  float thd = __builtin_inff(), thl = __builtin_inff(), tha = __builtin_inff();

  issue_tile(0, 0);
  int myidx = 0;

  for (int tile = 0; tile < NTILES; ++tile) {
    if (tile + 1 < NTILES) {
      wait_ds_0();                          // finish LDS reads before reuse
      issue_tile(tile + 1, (tile + 1) & 1); // prefetch next tile (8 issues)
      wait_async_le8();                     // current tile's 8 copies done
    } else {
      wait_async_0();
    }

    const float* lonp = wbase + (tile & 1) * (2 * TILE);
    const float* latp = lonp + TILE;
    const int tbase = tile * TILE + lane * 4;

#pragma unroll
    for (int j = 0; j < TILE / 128; ++j) {  // 4 x float4 per lane per tile
      const float4 L = *(const float4*)&lonp[j * 128 + lane * 4];
      const float4 A = *(const float4*)&latp[j * 128 + lane * 4];
      const int ib = tbase + j * 128;
      const float lo4[4] = {L.x, L.y, L.z, L.w};
      const float la4[4] = {A.x, A.y, A.z, A.w};
      float dd[4], al[4], aa[4];
#pragma unroll
      for (int c = 0; c < 4; ++c) {
        al[c] = __builtin_fabsf(lo4[c]);
        aa[c] = __builtin_fabsf(la4[c]);
        // exact IEEE RN mul/add/sqrt to match jnp's elementwise f32 ops
        dd[c] = __fsqrt_rn(__fadd_rn(__fmul_rn(lo4[c], lo4[c]),
                                     __fmul_rn(la4[c], la4[c])));
      }
      // '<=' keeps tie candidates so cross-lane index tie-break stays exact
      if ((dd[0] <= thd) | (dd[1] <= thd) | (dd[2] <= thd) | (dd[3] <= thd)) {
#pragma unroll
        for (int c = 0; c < 4; ++c) insert16(dd[c], ib + c, vd, jd);
      }
      if ((al[0] <= thl) | (al[1] <= thl) | (al[2] <= thl) | (al[3] <= thl)) {
#pragma unroll
        for (int c = 0; c < 4; ++c) insert16(al[c], ib + c, vl, jl);
      }
      if ((aa[0] <= tha) | (aa[1] <= tha) | (aa[2] <= tha) | (aa[3] <= tha)) {
#pragma unroll
        for (int c = 0; c < 4; ++c) insert16(aa[c], ib + c, va, ja);
      }
    }
    // refresh thresholds: min over lanes of each lane's 16th-best
    thd = wave_min32(vd[15]);
    thl = wave_min32(vl[15]);
    tha = wave_min32(va[15]);
  }

  // exact 32-way merge: 16 rounds of (value,index)-lex min reduction + pop
  auto merge_out = [&](float (&v)[16], int (&id)[16], size_t out_off,
                       bool record) {
    int wi[16];
#pragma unroll
    for (int r = 0; r < 16; ++r) {
      float bv = v[0];
      int   bi = id[0];
#pragma unroll
      for (int off = 16; off > 0; off >>= 1) {
        const float ov = __shfl_xor(bv, off, 32);
        const int   oi = __shfl_xor(bi, off, 32);
        if ((ov < bv) || ((ov == bv) && (oi < bi))) { bv = ov; bi = oi; }
      }
      if ((v[0] == bv) && (id[0] == bi)) {   // winner lane pops its head
#pragma unroll
        for (int q = 0; q < 15; ++q) { v[q] = v[q + 1]; id[q] = id[q + 1]; }
        v[15] = __builtin_inff(); id[15] = 0x7fffffff;
      }
      wi[r] = bi;
      if (record && ((lane & 15) == r)) myidx = bi; // lanes k, k+16 hold idx k
    }
    if (lane == 0) {
#pragma unroll
      for (int g = 0; g < 4; ++g) {
        float4 f = make_float4((float)wi[4 * g],     (float)wi[4 * g + 1],
                               (float)wi[4 * g + 2], (float)wi[4 * g + 3]);
        *(float4*)&out[out_off + (size_t)row * NH + g * 4] = f;
      }
    }
  };

  merge_out(vd, jd, IDD_OFF,   true);
  merge_out(vl, jl, IDLON_OFF, false);
  merge_out(va, ja, IDLAT_OFF, false);

  // gather: x_nearest[b][row][k] = x[b*S + idx_dist[row][k]]  (x lives in L2)
  const int k    = lane & 15;
  const int half = lane >> 4;
#pragma unroll
  for (int b2 = 0; b2 < B_DIM / 2; ++b2) {
    const int b = 2 * b2 + half;
    const float xv = x[(size_t)b * S_DIM + myidx];
    out[XN_OFF + (size_t)b * (T_DIM * NH) + (size_t)row * NH + k] = xv;
  }
}

extern "C" void kernel_launch(void* const* d_in, const int* in_sizes, int n_in,
                              void* d_out, int out_size, void* d_ws,
                              size_t ws_size, hipStream_t stream) {
  (void)in_sizes; (void)n_in; (void)out_size; (void)d_ws; (void)ws_size;
  const float* x    = (const float*)d_in[0];
  const float* dlon = (const float*)d_in[1];
  const float* dlat = (const float*)d_in[2];
  float* out = (float*)d_out;
  dim3 grid(T_DIM / WPB);  // 1024 blocks
  dim3 block(BLOCK);       // 256 threads = 8 wave32 (one row per wave)
  knn_topk_gather_gfx1250<<<grid, block, 0, stream>>>(x, dlon, dlat, out);
}